// CTCPrefixScorer_35656818491392
// MI455X (gfx1250) — compile-verified
//
#include <hip/hip_runtime.h>
#include <math.h>

// ---------------------------------------------------------------------------
// CTC prefix scorer, reformulated as a linear-domain GEMM for CDNA5 WMMA.
//
//   exp(log_psi_[h, o]) = sum_t  Q[h_row, t] * exp(x_masked[b, t, o])
//
// Q rows 0..7  : exp(r_prev[t,0,h]) + exp(r_prev[t,1,h])   (r_sum path)
// Q rows 8..15 : exp(r_prev[t,1,h])                        ("hit" path, o==last_id)
//
// The jax.lax.scan in the reference is dead code for the returned value
// (psi_stack only uses log_phi and r_pre), so it is not computed.
// x (160 MB) is streamed exactly once, coalesced  ->  ~7 us HBM bound
// (and x fits in the 192 MB L2 across graph replays).
// ---------------------------------------------------------------------------

#define LOGZERO -10000000000.0f

constexpr int Bn   = 8;       // batch
constexpr int Tn   = 500;     // time
constexpr int On   = 10000;   // vocab
constexpr int NBH  = 64;      // B * W hypotheses
constexpr int SNUM = 200;     // scoring ids per hyp
constexpr int KMAX = 512;     // padded K (k=0 reserved for output_length==0 term)
constexpr int MASK_STRIDE = 320;   // u32 words per hyp (10000 bits -> 313, pad 320)
constexpr int LDS_STRIDE  = 521;   // odd stride -> conflict-free column reads

typedef __attribute__((ext_vector_type(2))) float v2f;
typedef __attribute__((ext_vector_type(8))) float v8f;

// workspace layout (floats / u32):
constexpr size_t A_ELEMS = (size_t)Bn * 16 * KMAX;   //  64 K floats (256 KB)
constexpr size_t D_ELEMS = (size_t)Bn * 16 * On;     // 1.28 M floats (5.12 MB)
constexpr size_t M_ELEMS = (size_t)NBH * MASK_STRIDE;// 20480 u32   (80 KB)

__device__ __forceinline__ int imin(int a, int b) { return a < b ? a : b; }

__device__ __forceinline__ v8f wmma4(v2f a, v2f b, v8f c) {
    return __builtin_amdgcn_wmma_f32_16x16x4_f32(false, a, false, b,
                                                 (short)0, c, false, false);
}

// --- Kernel 1: build Q matrices (A operand) + zero membership bitmask -------
__global__ void ctc_prep_kernel(const float* __restrict__ r_prev,
                                const int* __restrict__ p_ol,
                                float* __restrict__ A,
                                unsigned* __restrict__ maskbits) {
    int tid = blockIdx.x * blockDim.x + threadIdx.x;
    int ol = *p_ol;
    int start = ol > 1 ? ol : 1;
    if (tid < (int)A_ELEMS) {
        int k = tid & (KMAX - 1);
        int m = (tid >> 9) & 15;
        int b = tid >> 13;
        int h = b * 8 + (m & 7);
        float v = 0.0f;
        if (k == 0) {
            // reserved slot: contributes p[b,0,o] iff output_length == 0
            v = (ol == 0) ? 1.0f : 0.0f;
        } else {
            int tau = start + k - 2;            // k=1 -> tau = start-1
            if (tau <= Tn - 2) {
                float r1 = r_prev[(tau * 2 + 1) * NBH + h];
                float e1 = __expf(r1);
                if (m < 8) {
                    float r0 = r_prev[(tau * 2 + 0) * NBH + h];
                    v = __expf(r0) + e1;        // exp(r_sum)
                } else {
                    v = e1;                     // exp(r_prev[:,1]) hit path
                }
            }
        }
        A[tid] = v;
    } else {
        int i = tid - (int)A_ELEMS;
        if (i < (int)M_ELEMS) maskbits[i] = 0u;
    }
}

// --- Kernel 2: scatter scoring-id membership bits ---------------------------
__global__ void ctc_scatter_kernel(const int* __restrict__ scoring_ids,
                                   unsigned* __restrict__ maskbits) {
    int tid = blockIdx.x * blockDim.x + threadIdx.x;
    if (tid >= NBH * SNUM) return;
    int h = tid / SNUM;
    int o = scoring_ids[tid];
    atomicOr(&maskbits[h * MASK_STRIDE + (o >> 5)], 1u << (o & 31));
}

// --- Kernel 3: D[b] = Q[b](16xK) @ exp(x_masked[b])(K x 10000) via WMMA -----
// Each wave owns TWO adjacent 16-wide o-tiles sharing the same A fragments.
// The single wave whose second tile is out of range aliases it onto the first
// tile (loads stay valid, result discarded) so the hot loop is branch-free.
__global__ void __launch_bounds__(256)
ctc_gemm_kernel(const float* __restrict__ x,
                const int* __restrict__ xlens,
                const int* __restrict__ p_ol,
                const float* __restrict__ A,
                float* __restrict__ D) {
    __shared__ float ldsA[16 * LDS_STRIDE];

    int b    = blockIdx.y;
    int tid  = threadIdx.x;
    int lane = tid & 31;
    int wave = tid >> 5;

    // stage A (16 x KMAX) into LDS, padded stride
    const float* Ab = A + (size_t)b * 16 * KMAX;
    for (int i = tid; i < 16 * KMAX; i += 256) {
        ldsA[(i >> 9) * LDS_STRIDE + (i & (KMAX - 1))] = Ab[i];
    }
    __syncthreads();

    int o0 = (blockIdx.x * 8 + wave) * 32;     // pair of 16-wide o-tiles
    if (o0 >= On) return;                      // wave-uniform: EXEC stays full
    bool has2 = (o0 + 16) < On;                // wave-uniform
    int  off2 = has2 ? 16 : 0;                 // alias 2nd tile onto 1st if OOB

    int ol    = *p_ol;
    int start = ol > 1 ? ol : 1;
    int xlen  = xlens[b];
    int K_live = xlen - start;                 // last live k (t = xlen-1); xlen <= Tn
    if (K_live < 0) K_live = 0;
    if (K_live > Tn - start) K_live = Tn - start;

    int n     = lane & 15;                     // A: row M, B: column N
    int khalf = (lane >> 4) << 1;              // lanes 16..31 handle K+2,K+3
    const float* ldsrow = ldsA + n * LDS_STRIDE;
    const float* xbase  = x + (size_t)b * Tn * On + (o0 + n);
    const float* xbase2 = xbase + off2;

    v8f c0 = {};
    v8f c1 = {};

    // ---- peel tile k0 = 0 (contains the reserved output_length==0 slot) ----
    {
        int kk = khalf;                        // 0 or 2
        v2f a;
        a.x = ldsrow[kk];
        a.y = ldsrow[kk + 1];
        int ta = start + kk - 1;               // unclamped t for row kk
        int tb = ta + 1;
        int t1 = (kk == 0) ? 0 : imin(ta, Tn - 1);
        int t2 = imin(tb, Tn - 1);
        float m1 = (kk == 0) ? 1.0f : ((ta < xlen) ? 1.0f : 0.0f);
        float m2 = (tb < xlen) ? 1.0f : 0.0f;
        float x1 = xbase[(size_t)t1 * On];
        float x2 = xbase[(size_t)t2 * On];
        float y1 = xbase2[(size_t)t1 * On];
        float y2 = xbase2[(size_t)t2 * On];
        v2f bf, bg;
        bf.x = __expf(x1) * m1;
        bf.y = __expf(x2) * m2;
        bg.x = __expf(y1) * m1;
        bg.y = __expf(y2) * m2;
        c0 = wmma4(a, bf, c0);
        c1 = wmma4(a, bg, c1);
    }

    // ---- main loop: all four k's live -> no clamp, no mask, streaming ptrs ----
    int k0 = 4;
    const float* pb  = xbase  + (size_t)(start + 3 + khalf) * On;
    const float* pb2 = xbase2 + (size_t)(start + 3 + khalf) * On;
#pragma unroll 4
    for (; k0 + 3 <= K_live; k0 += 4) {
        v2f a;
        a.x = ldsrow[k0 + khalf];
        a.y = ldsrow[k0 + khalf + 1];
        float x1 = pb[0];
        float x2 = pb[On];
        float y1 = pb2[0];
        float y2 = pb2[On];
        v2f bf, bg;
        bf.x = __expf(x1);
        bf.y = __expf(x2);
        bg.x = __expf(y1);
        bg.y = __expf(y2);
        c0 = wmma4(a, bf, c0);
        c1 = wmma4(a, bg, c1);
        pb  += (size_t)4 * On;
        pb2 += (size_t)4 * On;
    }

    // ---- tail: at most one tile, masked + clamped ----
    for (; k0 <= K_live; k0 += 4) {
        int kk = k0 + khalf;
        v2f a;
        a.x = ldsrow[kk];
        a.y = ldsrow[kk + 1];
        int ta = start + kk - 1;
        int tb = ta + 1;
        float m1 = (ta < xlen) ? 1.0f : 0.0f;
        float m2 = (tb < xlen) ? 1.0f : 0.0f;
        int t1 = imin(ta, Tn - 1);
        int t2 = imin(tb, Tn - 1);
        float x1 = xbase[(size_t)t1 * On];
        float x2 = xbase[(size_t)t2 * On];
        float y1 = xbase2[(size_t)t1 * On];
        float y2 = xbase2[(size_t)t2 * On];
        v2f bf, bg;
        bf.x = __expf(x1) * m1;
        bf.y = __expf(x2) * m2;
        bg.x = __expf(y1) * m1;
        bg.y = __expf(y2) * m2;
        c0 = wmma4(a, bf, c0);
        c1 = wmma4(a, bg, c1);
    }

    // C/D layout: VGPR r, lanes 0-15 -> (M=r, N=lane), lanes 16-31 -> (M=r+8)
    int mbase = (lane < 16) ? 0 : 8;
    float* Db = D + ((size_t)b * 16 + mbase) * On + (o0 + n);
#pragma unroll
    for (int r = 0; r < 8; ++r) Db[(size_t)r * On] = c0[r];
    if (has2) {
#pragma unroll
        for (int r = 0; r < 8; ++r) Db[(size_t)r * On + 16] = c1[r];
    }
}

// --- Kernel 4: epilogue: log, membership mask, EOS/BLANK, minus s_prev ------
__global__ void ctc_epilogue_kernel(const float* __restrict__ r_prev,
                                    const float* __restrict__ s_prev,
                                    const int* __restrict__ xlens,
                                    const int* __restrict__ last_ids,
                                    const float* __restrict__ D,
                                    const unsigned* __restrict__ maskbits,
                                    float* __restrict__ out) {
    int tid = blockIdx.x * blockDim.x + threadIdx.x;
    if (tid >= NBH * On) return;
    int h = tid / On;
    int o = tid - h * On;
    int b = h >> 3;
    float val;
    if (o == 0) {                              // BLANK
        val = LOGZERO;
    } else if (o == 2) {                       // EOS: r_sum at last frame
        int end = xlens[b] - 1;
        float a0 = r_prev[(end * 2 + 0) * NBH + h];
        float a1 = r_prev[(end * 2 + 1) * NBH + h];
        float mx = fmaxf(a0, a1);
        val = mx + logf(expf(a0 - mx) + expf(a1 - mx));
    } else {
        unsigned bit = (maskbits[h * MASK_STRIDE + (o >> 5)] >> (o & 31)) & 1u;
        if (bit) {
            int row = (o == last_ids[h]) ? 8 + (h & 7) : (h & 7);
            float d = D[((size_t)b * 16 + row) * On + o];
            val = (d > 0.0f) ? logf(d) : LOGZERO;
        } else {
            val = LOGZERO;
        }
    }
    out[tid] = val - s_prev[tid];
}

// ---------------------------------------------------------------------------
extern "C" void kernel_launch(void* const* d_in, const int* in_sizes, int n_in,
                              void* d_out, int out_size, void* d_ws, size_t ws_size,
                              hipStream_t stream) {
    const float* x           = (const float*)d_in[0];
    const float* r_prev      = (const float*)d_in[1];
    const float* s_prev      = (const float*)d_in[2];
    const int*   xlens       = (const int*)d_in[3];
    const int*   last_ids    = (const int*)d_in[4];
    const int*   scoring_ids = (const int*)d_in[5];
    const int*   p_ol        = (const int*)d_in[6];

    float*    A        = (float*)d_ws;
    float*    D        = A + A_ELEMS;
    unsigned* maskbits = (unsigned*)(D + D_ELEMS);
    float*    out      = (float*)d_out;

    int prepN = (int)(A_ELEMS + M_ELEMS);
    ctc_prep_kernel<<<(prepN + 255) / 256, 256, 0, stream>>>(r_prev, p_ol, A, maskbits);
    ctc_scatter_kernel<<<(NBH * SNUM + 255) / 256, 256, 0, stream>>>(scoring_ids, maskbits);
    // 625 o-tiles per batch, 2 tiles per wave, 8 waves per WG -> 40 blocks in x
    ctc_gemm_kernel<<<dim3(40, 8), 256, 0, stream>>>(x, xlens, p_ol, A, D);
    ctc_epilogue_kernel<<<(NBH * On + 255) / 256, 256, 0, stream>>>(
        r_prev, s_prev, xlens, last_ids, D, maskbits, out);
}